// Attention_2327872274829
// MI455X (gfx1250) — compile-verified
//
#include <hip/hip_runtime.h>
#include <hip/hip_bf16.h>
#include <math.h>

// ---------------------------------------------------------------------------
// Attention graph-conv block, fp32, MI455X (gfx1250, wave32).
// GEMMs via V_WMMA_F32_16X16X4_F32 (clause'd b64 A-loads, bias folded into
// the accumulator, scalar tail branches); gathers are b128 and rely on the
// 192MB L2 residency of the 160MB feature tables.
// ---------------------------------------------------------------------------

typedef float v2f __attribute__((ext_vector_type(2)));
typedef float v4f __attribute__((ext_vector_type(4)));
typedef float v8f __attribute__((ext_vector_type(8)));

#define C_CH   20
#define K_NBR  9
#define EPSBN  1e-5f
#define WPB    8          // waves per 256-thread block

// ---------------------------------------------------------------------------
// Kernel 1: Y = feature @ W1 + b1   ([N,20] x [20,20]), WMMA f32 16x16x4.
// One wave per 16-row tile; 5 K-steps; two 16-wide column tiles.
// ---------------------------------------------------------------------------
__global__ void k_gemm1(const float* __restrict__ F, const float* __restrict__ W1,
                        const float* __restrict__ b1, float* __restrict__ Y, int n)
{
    const int lane = threadIdx.x & 31;
    const int l16  = lane & 15;
    const int half = lane >> 4;                 // 0 -> k+0/1, 1 -> k+2/3
    const int wslot = __builtin_amdgcn_readfirstlane(threadIdx.x >> 5); // scalar wave slot
    const int wid  = blockIdx.x * WPB + wslot;  // scalar
    const int nw   = gridDim.x * WPB;           // scalar
    const int ntile = (n + 15) >> 4;

    // Preload B fragments (unconditional load + value select, no EXEC branches).
    v2f bfrag[2][5];
    #pragma unroll
    for (int t = 0; t < 2; ++t) {
        #pragma unroll
        for (int s = 0; s < 5; ++s) {
            const int col = t * 16 + l16;
            const int cc  = (col < C_CH) ? col : 0;
            const int k0  = 4 * s + 2 * half;
            const float bx = W1[k0 * C_CH + cc];
            const float by = W1[(k0 + 1) * C_CH + cc];
            v2f b;
            b.x = (col < C_CH) ? bx : 0.f;
            b.y = (col < C_CH) ? by : 0.f;
            bfrag[t][s] = b;
        }
    }

    // Bias per output column, folded into accumulator init.
    const int  col1  = 16 + l16;
    const bool c1ok  = (col1 < C_CH);
    const float bias0 = b1[l16];
    const float bias1c = b1[c1ok ? col1 : 0];
    const float bias1 = c1ok ? bias1c : 0.f;

    for (int tile = wid; tile < ntile; tile += nw) {     // tile is scalar
        const int row0 = tile << 4;                      // scalar
        const int row  = row0 + l16;
        const int rowc = (row < n) ? row : (n - 1);      // clamp: loads in-bounds
        const float* __restrict__ arow = F + (size_t)rowc * C_CH + 2 * half;

        v2f a[5];
        #pragma unroll
        for (int s = 0; s < 5; ++s)
            a[s] = *(const v2f*)(arow + 4 * s);          // global_load_b64

        v8f acc0, acc1;
        #pragma unroll
        for (int r = 0; r < 8; ++r) { acc0[r] = bias0; acc1[r] = bias1; }

        #pragma unroll
        for (int s = 0; s < 5; ++s) {
            acc0 = __builtin_amdgcn_wmma_f32_16x16x4_f32(false, a[s], false, bfrag[0][s],
                                                         (short)0, acc0, false, false);
            acc1 = __builtin_amdgcn_wmma_f32_16x16x4_f32(false, a[s], false, bfrag[1][s],
                                                         (short)0, acc1, false, false);
        }

        // C/D layout: VGPR r -> row (row0 + half*8 + r); col = l16 / 16+l16.
        const int rbase = row0 + half * 8;
        float* __restrict__ p0 = Y + (size_t)rbase * C_CH + l16;
        if (row0 + 16 <= n) {                            // scalar branch (tile uniform)
            #pragma unroll
            for (int r = 0; r < 8; ++r)
                p0[r * C_CH] = acc0[r];
            if (c1ok) {
                float* __restrict__ p1 = Y + (size_t)rbase * C_CH + col1;
                #pragma unroll
                for (int r = 0; r < 8; ++r)
                    p1[r * C_CH] = acc1[r];
            }
        } else {                                         // tail tile
            #pragma unroll
            for (int r = 0; r < 8; ++r) {
                if (rbase + r < n) {
                    p0[r * C_CH] = acc0[r];
                    if (c1ok)
                        Y[(size_t)(rbase + r) * C_CH + col1] = acc1[r];
                }
            }
        }
    }
}

// ---------------------------------------------------------------------------
// Per-wave partial channel sums / sums-of-squares over an [n,20] array.
// Deterministic: fixed row partition per wave, fixed accumulation order.
// ---------------------------------------------------------------------------
__global__ void k_stats(const float* __restrict__ src, float* __restrict__ part, int n)
{
    const int lane = threadIdx.x & 31;
    const int wslot = __builtin_amdgcn_readfirstlane(threadIdx.x >> 5);
    const int wid  = blockIdx.x * WPB + wslot;
    const int nw   = gridDim.x * WPB;
    if (lane < C_CH) {
        float s = 0.f, q = 0.f;
        for (int r = wid; r < n; r += nw) {
            const float v = src[(size_t)r * C_CH + lane];
            s += v;
            q += v * v;
        }
        part[wid * 2 * C_CH + lane]        = s;
        part[wid * 2 * C_CH + C_CH + lane] = q;
    }
}

// Reduce partials (fixed order) -> st[0..19]=gamma*rsqrt(var+eps), st[20..39]=beta-mean*scale
__global__ void k_finalize(const float* __restrict__ part, int nw, float nrows,
                           const float* __restrict__ g, const float* __restrict__ be,
                           float* __restrict__ st)
{
    const int c = threadIdx.x;
    if (c < C_CH) {
        float s = 0.f, q = 0.f;
        for (int w = 0; w < nw; ++w) {
            s += part[w * 2 * C_CH + c];
            q += part[w * 2 * C_CH + C_CH + c];
        }
        const float m   = s / nrows;
        const float var = q / nrows - m * m;
        const float sc  = g[c] * rsqrtf(var + EPSBN);
        st[c]        = sc;
        st[C_CH + c] = be[c] - m * sc;
    }
}

// ---------------------------------------------------------------------------
// Attention pass 1: x = bn1(y) on the fly; 9 neighbor dots (b128 gathers,
// L2-resident table), softmax over K=9, weighted sum -> O1, w.
// ---------------------------------------------------------------------------
__global__ void k_attn1(const float* __restrict__ Y, const float* __restrict__ st,
                        const int* __restrict__ idx, float* __restrict__ Wt,
                        float* __restrict__ O1, int n)
{
    const int r = blockIdx.x * blockDim.x + threadIdx.x;
    if (r >= n) return;

    const v4f* __restrict__ Yv = (const v4f*)Y;          // rows are 16B-aligned (80B)
    v4f* __restrict__ O1v = (v4f*)O1;

    v4f sc[5], sh[5];
    #pragma unroll
    for (int t = 0; t < 5; ++t) {
        sc[t] = *(const v4f*)(st + 4 * t);
        sh[t] = *(const v4f*)(st + C_CH + 4 * t);
    }

    v4f x[5];
    #pragma unroll
    for (int t = 0; t < 5; ++t)
        x[t] = Yv[(size_t)r * 5 + t] * sc[t] + sh[t];

    float d[K_NBR];
    float dmax = -3.0e38f;
    #pragma unroll
    for (int k = 0; k < K_NBR; ++k) {
        const int j = idx[(size_t)r * K_NBR + k];
        v4f accv = {};
        #pragma unroll
        for (int t = 0; t < 5; ++t) {
            const v4f p = Yv[(size_t)j * 5 + t] * sc[t] + sh[t];
            accv += x[t] * p;
        }
        const float acc = accv[0] + accv[1] + accv[2] + accv[3];
        d[k] = acc;
        dmax = fmaxf(dmax, acc);
    }
    float wsum = 0.f;
    #pragma unroll
    for (int k = 0; k < K_NBR; ++k) { d[k] = __expf(d[k] - dmax); wsum += d[k]; }
    const float inv = 1.f / wsum;

    v4f o[5] = {};
    #pragma unroll
    for (int k = 0; k < K_NBR; ++k) {
        const float wk = d[k] * inv;
        Wt[(size_t)r * K_NBR + k] = wk;
        const int j = idx[(size_t)r * K_NBR + k];
        #pragma unroll
        for (int t = 0; t < 5; ++t)
            o[t] += wk * (Yv[(size_t)j * 5 + t] * sc[t] + sh[t]);
    }
    #pragma unroll
    for (int t = 0; t < 5; ++t)
        O1v[(size_t)r * 5 + t] = o[t];
}

// Attention pass 2: O2[n] = sum_k w[n,k] * O1[idx[n,k]]  (b128 gathers)
__global__ void k_attn2(const float* __restrict__ O1, const float* __restrict__ Wt,
                        const int* __restrict__ idx, float* __restrict__ O2, int n)
{
    const int r = blockIdx.x * blockDim.x + threadIdx.x;
    if (r >= n) return;
    const v4f* __restrict__ O1v = (const v4f*)O1;
    v4f* __restrict__ O2v = (v4f*)O2;

    v4f o[5] = {};
    #pragma unroll
    for (int k = 0; k < K_NBR; ++k) {
        const float wk = Wt[(size_t)r * K_NBR + k];
        const int   j  = idx[(size_t)r * K_NBR + k];
        #pragma unroll
        for (int t = 0; t < 5; ++t)
            o[t] += wk * O1v[(size_t)j * 5 + t];
    }
    #pragma unroll
    for (int t = 0; t < 5; ++t)
        O2v[(size_t)r * 5 + t] = o[t];
}

// ---------------------------------------------------------------------------
// Kernel: U = concat(relu(bn2(O2)), feature) @ W3 + b3  ([N,40]x[40,20]).
// WMMA f32 16x16x4, 10 K-steps. Steps 0-4 source the bn2/relu half (k<20),
// steps 5-9 source feature (k>=20) -> zero per-element branching.
// ---------------------------------------------------------------------------
__global__ void k_gemm2(const float* __restrict__ O2, const float* __restrict__ st2,
                        const float* __restrict__ F, const float* __restrict__ W3,
                        const float* __restrict__ b3, float* __restrict__ U, int n)
{
    const int lane = threadIdx.x & 31;
    const int l16  = lane & 15;
    const int half = lane >> 4;
    const int wslot = __builtin_amdgcn_readfirstlane(threadIdx.x >> 5);
    const int wid  = blockIdx.x * WPB + wslot;
    const int nw   = gridDim.x * WPB;
    const int ntile = (n + 15) >> 4;

    v2f bfrag[2][10];
    #pragma unroll
    for (int t = 0; t < 2; ++t) {
        #pragma unroll
        for (int s = 0; s < 10; ++s) {
            const int col = t * 16 + l16;
            const int cc  = (col < C_CH) ? col : 0;
            const int k0  = 4 * s + 2 * half;
            const float bx = W3[k0 * C_CH + cc];
            const float by = W3[(k0 + 1) * C_CH + cc];
            v2f b;
            b.x = (col < C_CH) ? bx : 0.f;
            b.y = (col < C_CH) ? by : 0.f;
            bfrag[t][s] = b;
        }
    }

    // Affine coefficients for this lane's k-pairs (k0 = 4s+2*half, s=0..4)
    v2f asc[5], ash[5];
    #pragma unroll
    for (int s = 0; s < 5; ++s) {
        const int k0 = 4 * s + 2 * half;
        asc[s] = *(const v2f*)(st2 + k0);
        ash[s] = *(const v2f*)(st2 + C_CH + k0);
    }

    const int  col1  = 16 + l16;
    const bool c1ok  = (col1 < C_CH);
    const float bias0 = b3[l16];
    const float bias1c = b3[c1ok ? col1 : 0];
    const float bias1 = c1ok ? bias1c : 0.f;

    for (int tile = wid; tile < ntile; tile += nw) {
        const int row0 = tile << 4;
        const int row  = row0 + l16;
        const int rowc = (row < n) ? row : (n - 1);
        const float* __restrict__ orow = O2 + (size_t)rowc * C_CH + 2 * half;
        const float* __restrict__ frow = F  + (size_t)rowc * C_CH + 2 * half;

        v2f a[10];
        #pragma unroll
        for (int s = 0; s < 5; ++s) {                    // k = 0..19: relu(bn2(O2))
            v2f v = *(const v2f*)(orow + 4 * s);
            v = v * asc[s] + ash[s];
            v2f aa;
            aa.x = fmaxf(v.x, 0.f);
            aa.y = fmaxf(v.y, 0.f);
            a[s] = aa;
        }
        #pragma unroll
        for (int s = 0; s < 5; ++s)                      // k = 20..39: feature
            a[5 + s] = *(const v2f*)(frow + 4 * s);

        v8f acc0, acc1;
        #pragma unroll
        for (int r = 0; r < 8; ++r) { acc0[r] = bias0; acc1[r] = bias1; }

        #pragma unroll
        for (int s = 0; s < 10; ++s) {
            acc0 = __builtin_amdgcn_wmma_f32_16x16x4_f32(false, a[s], false, bfrag[0][s],
                                                         (short)0, acc0, false, false);
            acc1 = __builtin_amdgcn_wmma_f32_16x16x4_f32(false, a[s], false, bfrag[1][s],
                                                         (short)0, acc1, false, false);
        }

        const int rbase = row0 + half * 8;
        float* __restrict__ p0 = U + (size_t)rbase * C_CH + l16;
        if (row0 + 16 <= n) {
            #pragma unroll
            for (int r = 0; r < 8; ++r)
                p0[r * C_CH] = acc0[r];
            if (c1ok) {
                float* __restrict__ p1 = U + (size_t)rbase * C_CH + col1;
                #pragma unroll
                for (int r = 0; r < 8; ++r)
                    p1[r * C_CH] = acc1[r];
            }
        } else {
            #pragma unroll
            for (int r = 0; r < 8; ++r) {
                if (rbase + r < n) {
                    p0[r * C_CH] = acc0[r];
                    if (c1ok)
                        U[(size_t)(rbase + r) * C_CH + col1] = acc1[r];
                }
            }
        }
    }
}

// Final in-place BN3 affine + ReLU over d_out ([N,20]), vec4.
__global__ void k_final(float* __restrict__ U, const float* __restrict__ st, long long total4)
{
    const long long i = (long long)blockIdx.x * blockDim.x + threadIdx.x;
    if (i >= total4) return;
    v4f* Uv = (v4f*)U;
    const int c0 = (int)((i * 4) % C_CH);                // in {0,4,8,12,16}
    const v4f sc = *(const v4f*)(st + c0);
    const v4f sh = *(const v4f*)(st + C_CH + c0);
    v4f v = Uv[i] * sc + sh;
    #pragma unroll
    for (int t = 0; t < 4; ++t)
        v[t] = fmaxf(v[t], 0.f);
    Uv[i] = v;
}

// ---------------------------------------------------------------------------
// Host-side launch
// ---------------------------------------------------------------------------
extern "C" void kernel_launch(void* const* d_in, const int* in_sizes, int n_in,
                              void* d_out, int out_size, void* d_ws, size_t ws_size,
                              hipStream_t stream)
{
    const float* F   = (const float*)d_in[0];
    const int*   idx = (const int*)  d_in[1];
    const float* W1  = (const float*)d_in[2];
    const float* b1  = (const float*)d_in[3];
    const float* g1  = (const float*)d_in[4];
    const float* be1 = (const float*)d_in[5];
    const float* g2  = (const float*)d_in[6];
    const float* be2 = (const float*)d_in[7];
    const float* W3  = (const float*)d_in[8];
    const float* b3  = (const float*)d_in[9];
    const float* g3  = (const float*)d_in[10];
    const float* be3 = (const float*)d_in[11];

    const int N = in_sizes[0] / C_CH;

    const size_t szRow = (size_t)N * C_CH * sizeof(float);  // [N,20]
    const size_t szW   = (size_t)N * K_NBR * sizeof(float); // [N,9]
    char* ws = (char*)d_ws;
    float* Y    = (float*)(ws);                    // [N,20]  (reused as O2 later)
    float* O1   = (float*)(ws + szRow);            // [N,20]
    float* Wt   = (float*)(ws + 2 * szRow);        // [N,9]
    float* part = (float*)(ws + 2 * szRow + szW);  // [NW, 40]
    const int STAT_BLOCKS = 512;
    const int NW = STAT_BLOCKS * WPB;
    float* st1  = part + (size_t)NW * 2 * C_CH;    // [40]
    float* st2  = st1 + 2 * C_CH;
    float* st3  = st2 + 2 * C_CH;
    float* O2   = Y;                               // Y dead after k_attn1
    float* U    = (float*)d_out;                   // pre-BN3 result lives in d_out

    const int TPB = 256;
    const int GEMM_BLOCKS = 1024;                  // 8192 waves over 16-row tiles
    const int rowBlocks = (N + TPB - 1) / TPB;
    const long long total4 = (long long)N * (C_CH / 4);
    const int elemBlocks = (int)((total4 + TPB - 1) / TPB);

    k_gemm1<<<GEMM_BLOCKS, TPB, 0, stream>>>(F, W1, b1, Y, N);
    k_stats<<<STAT_BLOCKS, TPB, 0, stream>>>(Y, part, N);
    k_finalize<<<1, 32, 0, stream>>>(part, NW, (float)N, g1, be1, st1);
    k_attn1<<<rowBlocks, TPB, 0, stream>>>(Y, st1, idx, Wt, O1, N);
    k_attn2<<<rowBlocks, TPB, 0, stream>>>(O1, Wt, idx, O2, N);
    k_stats<<<STAT_BLOCKS, TPB, 0, stream>>>(O2, part, N);
    k_finalize<<<1, 32, 0, stream>>>(part, NW, (float)N, g2, be2, st2);
    k_gemm2<<<GEMM_BLOCKS, TPB, 0, stream>>>(O2, st2, F, W3, b3, U, N);
    k_stats<<<STAT_BLOCKS, TPB, 0, stream>>>(U, part, N);
    k_finalize<<<1, 32, 0, stream>>>(part, NW, (float)N, g3, be3, st3);
    k_final<<<elemBlocks, TPB, 0, stream>>>(U, st3, total4);
}